// OptimizeSNR_71502615544035
// MI455X (gfx1250) — compile-verified
//
#include <hip/hip_runtime.h>

typedef __attribute__((ext_vector_type(2))) float v2f;
typedef __attribute__((ext_vector_type(8))) float v8f;

#define BB     8
#define CC     128
#define LL     32768
#define KF     129        // filter taps
#define PADK   64         // KF/2
#define NSTEPS 64         // K-loop steps of K=4 -> K' = 256 (taps 241..255 are zero)
#define KPS    260        // W column stride (dwords): 260%64=4 -> conflict-free, even -> 8B aligned
#define CHUNK  4096       // outputs per workgroup
#define TILE   128        // outputs per wave-tile
#define TPW    4          // tiles per wave
#define NSTAGE 4240       // x elements staged (max index used: 4238)
#define NSX    4256

__global__ __launch_bounds__(256) void snr_wmma_kernel(
    const float* __restrict__ x,
    const float* __restrict__ wm,
    const float* __restrict__ wa,
    const float* __restrict__ kr,
    const float* __restrict__ ki,
    float* __restrict__ out)
{
    __shared__ float xs[NSX];
    __shared__ float wlds[16 * KPS];     // TRANSPOSED: wlds[n*KPS + kk]
    __shared__ float red[128];
    __shared__ float s_cr, s_ci;

    const int tid   = threadIdx.x;
    const int bid   = blockIdx.x;
    const int chunk = bid & 7;           // L / CHUNK = 8 chunks per row
    const int row   = bid >> 3;          // b*CC + c
    const int c     = row & (CC - 1);
    const int chunkBase = chunk * CHUNK;

    // ---- stage shifted-filter matrix, transposed: column n = 2*s + f ----
    for (int i = tid; i < 16 * KPS; i += 256) {
        int n  = i / KPS;
        int kk = i - n * KPS;
        int sh = n >> 1;
        int im = n & 1;
        int k  = kk - (sh << 4);
        float v = 0.0f;
        if ((unsigned)k < (unsigned)KF) v = im ? -ki[k] : kr[k];
        wlds[i] = v;
    }

    // ---- stage x window (float4 fast path; scalar + zero-pad at edges) ----
    const float* xrow = x + (size_t)row * LL;
    const int x0 = chunkBase - PADK;     // chunkBase 4096-aligned -> in-range l is 4-aligned
    for (int i4 = tid * 4; i4 < NSTAGE; i4 += 1024) {
        int l = x0 + i4;
        if (l >= 0 && (l + 3) < LL && (i4 + 3) < NSTAGE) {
            *(float4*)(xs + i4) = *(const float4*)(xrow + l);
        } else {
#pragma unroll
            for (int u = 0; u < 4; ++u) {
                int ii = i4 + u;
                if (ii < NSTAGE) {
                    int lu = l + u;
                    xs[ii] = ((unsigned)lu < (unsigned)LL) ? xrow[lu] : 0.0f;
                }
            }
        }
    }

    // ---- per-channel complex scale: nm = B*softmax(wm)[c]; phase = atan(wa[c]) ----
    float wmv = (tid < CC) ? wm[tid] : -3.4e38f;
    if (tid < 128) red[tid] = wmv;
    __syncthreads();
    for (int st = 64; st >= 1; st >>= 1) {
        if (tid < st) red[tid] = fmaxf(red[tid], red[tid + st]);
        __syncthreads();
    }
    float mx = red[0];
    __syncthreads();
    if (tid < 128) red[tid] = expf(wmv - mx);
    __syncthreads();
    for (int st = 64; st >= 1; st >>= 1) {
        if (tid < st) red[tid] += red[tid + st];
        __syncthreads();
    }
    if (tid == 0) {
        float nm  = (float)BB * expf(wm[c] - mx) / red[0];
        float t   = wa[c];
        float inv = rsqrtf(1.0f + t * t);   // cos(atan t); sin = t * inv
        s_cr = nm * inv;
        s_ci = nm * t * inv;
    }
    __syncthreads();

    // ---- WMMA main: each wave computes TPW tiles of 128 outputs ----
    const int wave = tid >> 5;
    const int lane = tid & 31;
    const int j    = lane & 15;           // A: row M ; B/C/D: col N
    const int half = lane >> 4;
    const int h2   = half << 1;           // K sub-offset for upper half-wave
    const int sh   = j >> 1;              // shift index
    const int im   = j & 1;               // 0 = real col, 1 = imag col
    const float cr = s_cr;
    const float ci = s_ci;
    const float* wrow = wlds + j * KPS + h2;  // 8B-aligned
    float* orow = out + (size_t)row * (size_t)(2 * LL);

    for (int t = 0; t < TPW; ++t) {
        const int tt = wave * TPW + t;
        const int xo = tt * TILE + j + h2;

        v8f acc0 = {0.f, 0.f, 0.f, 0.f, 0.f, 0.f, 0.f, 0.f};
        v8f acc1 = {0.f, 0.f, 0.f, 0.f, 0.f, 0.f, 0.f, 0.f};
        v2f a[2][8];

        // prologue: batch 0
#pragma unroll
        for (int u = 0; u < 8; ++u) {
            int o = u * 4;
            a[0][u].x = xs[xo + o];
            a[0][u].y = xs[xo + o + 1];
        }

#pragma unroll
        for (int sb = 0; sb < 8; ++sb) {
            const int cur = sb & 1;
            const int nxt = cur ^ 1;
            if (sb < 7) {                 // prefetch next batch of 8 A operands
#pragma unroll
                for (int u = 0; u < 8; ++u) {
                    int o = (sb + 1) * 32 + u * 4;
                    a[nxt][u].x = xs[xo + o];
                    a[nxt][u].y = xs[xo + o + 1];
                }
            }
#pragma unroll
            for (int u = 0; u < 8; ++u) {
                v2f w = *(const v2f*)(wrow + sb * 32 + u * 4);  // ds_load_b64 imm offset
                if (u & 1)
                    acc1 = __builtin_amdgcn_wmma_f32_16x16x4_f32(
                        false, a[cur][u], false, w, (short)0, acc1, false, false);
                else
                    acc0 = __builtin_amdgcn_wmma_f32_16x16x4_f32(
                        false, a[cur][u], false, w, (short)0, acc0, false, false);
            }
        }

        // ---- epilogue: lane pair (2s,2s+1) shares output position l ----
        const int lbase = chunkBase + tt * TILE + (half << 3) + (sh << 4);
        float rr[8], qq[8];
#pragma unroll
        for (int v = 0; v < 8; ++v) {
            float mine = acc0[v] + acc1[v];
            float part = __shfl_xor(mine, 1, 32);
            float yr = im ? part : mine;
            float yi = im ? mine : part;
            rr[v] = __fmaf_rn(cr, yr,  ci * yi);   // real
            qq[v] = __fmaf_rn(cr, yi, -ci * yr);   // imag
        }
        const int vb = im ? 4 : 0;                 // even lane: l..l+3, odd lane: l+4..l+7
        const int l0 = lbase + vb;
        float4 st0 = {rr[vb],     qq[vb],     rr[vb + 1], qq[vb + 1]};
        float4 st1 = {rr[vb + 2], qq[vb + 2], rr[vb + 3], qq[vb + 3]};
        *(float4*)(orow + 2 * l0)     = st0;       // 16B aligned
        *(float4*)(orow + 2 * l0 + 4) = st1;
    }
}

extern "C" void kernel_launch(void* const* d_in, const int* in_sizes, int n_in,
                              void* d_out, int out_size, void* d_ws, size_t ws_size,
                              hipStream_t stream) {
    const float* x  = (const float*)d_in[0];
    const float* wm = (const float*)d_in[1];
    const float* wa = (const float*)d_in[2];
    const float* kr = (const float*)d_in[3];
    const float* ki = (const float*)d_in[4];
    float* out = (float*)d_out;

    dim3 grid(BB * CC * (LL / CHUNK));   // 8192 workgroups
    dim3 block(256);                     // 8 wave32 per workgroup
    snr_wmma_kernel<<<grid, block, 0, stream>>>(x, wm, wa, kr, ki, out);
}